// NLM_7962869366897
// MI455X (gfx1250) — compile-verified
//
#include <hip/hip_runtime.h>

#define B_SZ 512
#define D_SZ 2048
#define M_SZ 32
#define N2H  128   // 2*H

typedef __attribute__((ext_vector_type(2))) float v2f;
typedef __attribute__((ext_vector_type(8))) float v8f;

// W pairs: 16 pair-rows (pi = ks*2 + khalf/2), 128 cols used, stride 144 float2
//   -> half-wave row offset = 144*2 mod 64 banks = 32: disjoint bank groups.
#define WPS 144
// A pairs: 128 rows x 17 float2 (34 floats, even -> aligned b64 loads,
//   34*m mod 64 distinct for m=0..15 -> conflict-free).
#define APS 17

__global__ __launch_bounds__(256)
void ctm_superlinear_kernel(const float* __restrict__ st,    // [B, D, M]
                            const float* __restrict__ w1a,   // [M, 2H, D]
                            const float* __restrict__ b1a,   // [1, D, 2H]
                            const float* __restrict__ Ta,    // [1]
                            const float* __restrict__ w1b,   // [H, 2, D]
                            const float* __restrict__ b1b,   // [1, D, 2]
                            const float* __restrict__ Tb,    // [1]
                            float* __restrict__ out)         // [B, D]
{
    __shared__ v2f  Wp[16 * WPS];        // 18432 B, fragment-packed w1a[:,:,d]
    __shared__ v2f  A2[128 * APS];       // 17408 B, fragment-packed A chunk
    __shared__ float bias_lds[N2H];
    __shared__ float w1b_lds[N2H];       // index h*2+j

    const int d    = blockIdx.x;
    const int t    = threadIdx.x;
    const int lane = t & 31;
    const int wv   = t >> 5;        // wave 0..7
    const int half = lane >> 4;     // 0 | 1
    const int l16  = lane & 15;

    float* Wf = (float*)Wp;
    float* Af = (float*)A2;

    // ---- stage w1a[:, :, d] packed as K-pairs: element (m,n) at (m*128+n)*D + d
    //      pair-row pi = (m>>2)*2 + ((m>>1)&1), component m&1
    #pragma unroll
    for (int i = 0; i < 16; ++i) {
        int idx = t + i * 256;              // 0..4095 = m*128+n
        int m = idx >> 7;
        int n = idx & 127;
        int pi = ((m >> 2) << 1) | ((m >> 1) & 1);
        Wf[(pi * WPS + n) * 2 + (m & 1)] = w1a[(size_t)idx * D_SZ + d];
    }
    if (t < N2H) {
        bias_lds[t] = b1a[d * N2H + t];
        w1b_lds[t]  = w1b[(size_t)t * D_SZ + d];   // t = h*2+j
    }
    const float invTa = 1.0f / Ta[0];
    const float invTb = 1.0f / Tb[0];
    const float bb0 = b1b[d * 2 + 0];
    const float bb1 = b1b[d * 2 + 1];
    __syncthreads();

    for (int chunk = 0; chunk < 4; ++chunk) {
        const int b0 = chunk * 128;

        // ---- stage A chunk (non-temporal: stream st once, keep w1a in L2)
        #pragma unroll
        for (int r = 0; r < 16; ++r) {
            int row = wv * 16 + r;
            Af[row * (2 * APS) + lane] = __builtin_nontemporal_load(
                &st[(size_t)(b0 + row) * (D_SZ * M_SZ) + (size_t)d * M_SZ + lane]);
        }
        __syncthreads();

        // prefetch next chunk's A rows (global_prefetch_b8)
        if (chunk < 3 && t < 128) {
            __builtin_prefetch(
                &st[(size_t)(b0 + 128 + t) * (D_SZ * M_SZ) + (size_t)d * M_SZ], 0, 1);
        }

        // ---- A fragments: one aligned ds_load_b64 each
        v2f af[8];
        const int arow = (wv * 16 + l16) * APS;
        #pragma unroll
        for (int ks = 0; ks < 8; ++ks)
            af[ks] = A2[arow + ks * 2 + half];

        float p0[8], p1[8];
        #pragma unroll
        for (int r = 0; r < 8; ++r) { p0[r] = 0.0f; p1[r] = 0.0f; }

        // ---- 4 GLU column-pair groups: cols g*16+l16 (value) and +64 (gate)
        #pragma unroll
        for (int g = 0; g < 4; ++g) {
            v8f accA = {};
            v8f accB = {};
            #pragma unroll
            for (int ks = 0; ks < 8; ++ks) {
                const int prow = (ks * 2 + half) * WPS;
                v2f bfA = Wp[prow + g * 16 + l16];
                v2f bfB = Wp[prow + g * 16 + 64 + l16];
                accA = __builtin_amdgcn_wmma_f32_16x16x4_f32(
                    false, af[ks], false, bfA, (short)0, accA, false, false);
                accB = __builtin_amdgcn_wmma_f32_16x16x4_f32(
                    false, af[ks], false, bfB, (short)0, accB, false, false);
            }
            // GLU (gate col h+64 lives in the same lane as h), fold stage-2 weights
            const int h = g * 16 + l16;
            const float biasA = bias_lds[h];
            const float biasB = bias_lds[h + 64];
            const float w0 = w1b_lds[h * 2 + 0];
            const float w1 = w1b_lds[h * 2 + 1];
            #pragma unroll
            for (int r = 0; r < 8; ++r) {
                float a = (accA[r] + biasA) * invTa;
                float b = (accB[r] + biasB) * invTa;
                float y = a * (1.0f / (1.0f + __expf(-b)));
                p0[r] = fmaf(y, w0, p0[r]);
                p1[r] = fmaf(y, w1, p1[r]);
            }
        }

        // ---- reduce stage-2 partials across the 16 lanes of each half-wave
        #pragma unroll
        for (int off = 1; off < 16; off <<= 1) {
            #pragma unroll
            for (int r = 0; r < 8; ++r) {
                p0[r] += __shfl_xor(p0[r], off, 32);
                p1[r] += __shfl_xor(p1[r], off, 32);
            }
        }

        // ---- final GLU + store: lane l16==r of each half writes row r+8*half
        #pragma unroll
        for (int r = 0; r < 8; ++r) {
            if (l16 == r) {
                float z0 = (p0[r] + bb0) * invTb;
                float z1 = (p1[r] + bb1) * invTb;
                float o  = z0 * (1.0f / (1.0f + __expf(-z1)));
                int brow = b0 + wv * 16 + r + 8 * half;
                out[(size_t)brow * D_SZ + d] = o;
            }
        }
        __syncthreads();   // protect A2 before next chunk refill
    }
}

extern "C" void kernel_launch(void* const* d_in, const int* in_sizes, int n_in,
                              void* d_out, int out_size, void* d_ws, size_t ws_size,
                              hipStream_t stream) {
    const float* st  = (const float*)d_in[0];
    const float* w1a = (const float*)d_in[1];
    const float* b1a = (const float*)d_in[2];
    const float* Ta  = (const float*)d_in[3];
    const float* w1b = (const float*)d_in[4];
    const float* b1b = (const float*)d_in[5];
    const float* Tb  = (const float*)d_in[6];
    float* out = (float*)d_out;

    dim3 grid(D_SZ, 1, 1);
    dim3 block(256, 1, 1);
    ctm_superlinear_kernel<<<grid, block, 0, stream>>>(st, w1a, b1a, Ta, w1b, b1b, Tb, out);
}